// LocationSensitiveAttention_8169027797353
// MI455X (gfx1250) — compile-verified
//
#include <hip/hip_runtime.h>
#include <hip/hip_bf16.h>
#include <math.h>

// Problem constants (from reference)
#define B_    128
#define T_    1024
#define RNN_  1024
#define EMB_  512
#define ATT_  128
#define NF_   32
#define KS_   31
#define PADK_ 15

typedef __attribute__((ext_vector_type(2))) float v2f;
typedef __attribute__((ext_vector_type(8))) float v8f;

// ---------------------------------------------------------------------------
// One 16x16 f32 output tile via V_WMMA_F32_16X16X4_F32.
//   C[m0:m0+16, n0:n0+16] = A[m0:m0+16, 0:K] * Bm[0:K, n0:n0+16]
// A-frag (16x4 f32): lane l holds A[m0 + (l&15)][k + 2*(l>>4) + {0,1}]
// B-frag (4x16 f32): lane l holds B[k + 2*(l>>4) + {0,1}][n0 + (l&15)]
// C/D (16x16 f32):  vgpr r, lane l -> row m0 + r + 8*(l>>4), col n0 + (l&15)
// ---------------------------------------------------------------------------
__device__ __forceinline__ v8f wmma_f32_tile(const float* __restrict__ A, int lda,
                                             const float* __restrict__ Bm, int ldb,
                                             int K, int m0, int n0) {
  const int lane = threadIdx.x & 31;
  const int half = lane >> 4;
  const int lidx = lane & 15;
  v8f acc = {0.f, 0.f, 0.f, 0.f, 0.f, 0.f, 0.f, 0.f};
  const float* arow = A + (size_t)(m0 + lidx) * lda + 2 * half;
  const float* bcol = Bm + (size_t)(2 * half) * ldb + n0 + lidx;
  for (int k = 0; k < K; k += 4) {
    v2f a, b;
    a.x = arow[k];
    a.y = arow[k + 1];
    b.x = bcol[(size_t)k * ldb];
    b.y = bcol[(size_t)(k + 1) * ldb];
    acc = __builtin_amdgcn_wmma_f32_16x16x4_f32(false, a, false, b,
                                                (short)0, acc, false, false);
  }
  return acc;
}

// ---------------------------------------------------------------------------
// Kernel 1: pq = hidden(128x1024) @ Wq(1024x128)   -> pq (128x128)
// grid(2,8), block 128 (4 waves); wave -> (m-tile = by, n-tile = bx*4+wave)
// ---------------------------------------------------------------------------
__global__ void pq_wmma_kernel(const float* __restrict__ hidden,
                               const float* __restrict__ Wq,
                               float* __restrict__ pq) {
  const int wave = threadIdx.x >> 5;
  const int n0 = (blockIdx.x * 4 + wave) * 16;
  const int m0 = blockIdx.y * 16;
  v8f acc = wmma_f32_tile(hidden, RNN_, Wq, ATT_, RNN_, m0, n0);
  const int lane = threadIdx.x & 31;
  const int half = lane >> 4, lidx = lane & 15;
#pragma unroll
  for (int r = 0; r < 8; ++r) {
    pq[(m0 + r + 8 * half) * ATT_ + n0 + lidx] = acc[r];
  }
}

// ---------------------------------------------------------------------------
// Kernel 2: conv1d SAME over T, 2 in-channels, 32 filters.
// Writes loc as (B, T, 32) so it is directly the WMMA A-matrix (row=t, K=f).
// grid(T/256, B), block 256
// ---------------------------------------------------------------------------
__global__ void conv_loc_kernel(const float* __restrict__ awc,   // (B,2,T)
                                const float* __restrict__ Wc,    // (32,2,31)
                                float* __restrict__ loc) {       // (B,T,32)
  const int b = blockIdx.y;
  const int t0 = blockIdx.x * 256;
  const int tid = threadIdx.x;

  __shared__ float s_in[2][256 + KS_ - 1];
  __shared__ float s_w[NF_ * 2 * KS_];

  for (int i = tid; i < NF_ * 2 * KS_; i += 256) s_w[i] = Wc[i];
#pragma unroll
  for (int c = 0; c < 2; ++c) {
    for (int i = tid; i < 256 + KS_ - 1; i += 256) {
      const int t = t0 - PADK_ + i;
      s_in[c][i] = (t >= 0 && t < T_) ? awc[((size_t)b * 2 + c) * T_ + t] : 0.f;
    }
  }
  __syncthreads();

  const int t = t0 + tid;
  float* dst = loc + ((size_t)b * T_ + t) * NF_;
  for (int f = 0; f < NF_; ++f) {
    const float* w0 = &s_w[(f * 2 + 0) * KS_];
    const float* w1 = &s_w[(f * 2 + 1) * KS_];
    float acc = 0.f;
#pragma unroll
    for (int k = 0; k < KS_; ++k) {
      acc = fmaf(s_in[0][tid + k], w0[k], acc);
      acc = fmaf(s_in[1][tid + k], w1[k], acc);
    }
    dst[f] = acc;
  }
}

// ---------------------------------------------------------------------------
// Kernel 3: fused energies. One wave owns a (b, 16-t tile):
//   - loads the 8 A-fragments (loc, K=32) ONCE into registers
//   - loops over all 8 a-tiles of W_loc_proj: 8 WMMAs each (64 total/wave)
//   - epilogue per tile: + pq + processed_memory -> tanh -> *v, accumulated
//     in registers across a-tiles
//   - 16-lane shfl_xor reduction over a, plain store of energies (no atomics)
// grid(T/64, B), block 128 (4 waves); wave -> t-tile = bx*4+wave
// ---------------------------------------------------------------------------
__global__ void energy_wmma_kernel(const float* __restrict__ loc,  // (B,T,32)
                                   const float* __restrict__ Wlp,  // (32,128)
                                   const float* __restrict__ pq,   // (B,128)
                                   const float* __restrict__ pm,   // (B,T,128)
                                   const float* __restrict__ v,    // (128)
                                   float* __restrict__ energies) { // (B,T)
  const int b = blockIdx.y;
  const int wave = threadIdx.x >> 5;
  const int t0 = (blockIdx.x * 4 + wave) * 16;
  const int lane = threadIdx.x & 31;
  const int half = lane >> 4, lidx = lane & 15;

  // A-fragments for rows t0..t0+15, K = 0..31 (loaded once, reused 8x)
  const float* arow = loc + ((size_t)b * T_ + t0 + lidx) * NF_ + 2 * half;
  v2f afrag[8];
#pragma unroll
  for (int kk = 0; kk < 8; ++kk) {
    afrag[kk].x = arow[4 * kk];
    afrag[kk].y = arow[4 * kk + 1];
  }

  const float* pmb = pm + (size_t)b * T_ * ATT_;
  float esum[8];
#pragma unroll
  for (int r = 0; r < 8; ++r) esum[r] = 0.f;

  for (int n0 = 0; n0 < ATT_; n0 += 16) {
    const float* bcol = Wlp + (2 * half) * ATT_ + n0 + lidx;
    v8f acc = {0.f, 0.f, 0.f, 0.f, 0.f, 0.f, 0.f, 0.f};
#pragma unroll
    for (int kk = 0; kk < 8; ++kk) {
      v2f bfrag;
      bfrag.x = bcol[(size_t)(4 * kk) * ATT_];
      bfrag.y = bcol[(size_t)(4 * kk + 1) * ATT_];
      acc = __builtin_amdgcn_wmma_f32_16x16x4_f32(false, afrag[kk], false, bfrag,
                                                  (short)0, acc, false, false);
    }
    const int a = n0 + lidx;
    const float pqv = pq[b * ATT_ + a];
    const float vv = v[a];
#pragma unroll
    for (int r = 0; r < 8; ++r) {
      const int t = t0 + r + 8 * half;
      float x = acc[r] + pqv + pmb[(size_t)t * ATT_ + a];
      esum[r] = fmaf(tanhf(x), vv, esum[r]);
    }
  }

  // reduce over the 16 lanes of each half-wave (the a dimension)
#pragma unroll
  for (int r = 0; r < 8; ++r) {
    float e = esum[r];
    e += __shfl_xor(e, 1, 32);
    e += __shfl_xor(e, 2, 32);
    e += __shfl_xor(e, 4, 32);
    e += __shfl_xor(e, 8, 32);
    if (lidx == 0) energies[b * T_ + t0 + r + 8 * half] = e;
  }
}

// ---------------------------------------------------------------------------
// Kernel 4: masked softmax over T per batch row. grid(B), block 256 (4 t each)
// ---------------------------------------------------------------------------
__global__ void softmax_kernel(const float* __restrict__ energies,
                               const unsigned char* __restrict__ mask,
                               float* __restrict__ weights) {
  const int b = blockIdx.x;
  const int tid = threadIdx.x;
  __shared__ float red[256];

  float e[4];
  float mx = -INFINITY;
#pragma unroll
  for (int i = 0; i < 4; ++i) {
    const int t = tid + i * 256;
    float x = energies[b * T_ + t];
    if (mask[b * T_ + t]) x = -INFINITY;
    e[i] = x;
    mx = fmaxf(mx, x);
  }
  red[tid] = mx;
  __syncthreads();
  for (int s = 128; s > 0; s >>= 1) {
    if (tid < s) red[tid] = fmaxf(red[tid], red[tid + s]);
    __syncthreads();
  }
  mx = red[0];
  __syncthreads();

  float sum = 0.f;
#pragma unroll
  for (int i = 0; i < 4; ++i) {
    e[i] = __expf(e[i] - mx);
    sum += e[i];
  }
  red[tid] = sum;
  __syncthreads();
  for (int s = 128; s > 0; s >>= 1) {
    if (tid < s) red[tid] += red[tid + s];
    __syncthreads();
  }
  const float inv = 1.f / red[0];
#pragma unroll
  for (int i = 0; i < 4; ++i) weights[b * T_ + tid + i * 256] = e[i] * inv;
}

// ---------------------------------------------------------------------------
// Kernel 5: context[b,d] = sum_t w[b,t] * memory[b,t,d]. Streams the 256 MB
// memory tensor once with b64 (float2) loads; weights staged through LDS.
// grid(EMB/256, B), block 128 (each lane owns 2 consecutive d's).
// ---------------------------------------------------------------------------
__global__ void context_kernel(const float* __restrict__ weights, // (B,T)
                               const float* __restrict__ memory,  // (B,T,EMB)
                               float* __restrict__ ctx) {         // (B,EMB)
  const int b = blockIdx.y;
  const int d = (blockIdx.x * 128 + threadIdx.x) * 2;
  __shared__ float s_w[256];

  const float2* memb = (const float2*)(memory + (size_t)b * T_ * EMB_ + d);
  const float* wb = weights + b * T_;
  float2 acc = {0.f, 0.f};

  for (int tc = 0; tc < T_; tc += 256) {
    __syncthreads();
    s_w[threadIdx.x] = wb[tc + threadIdx.x];
    s_w[threadIdx.x + 128] = wb[tc + threadIdx.x + 128];
    __syncthreads();
    if (tc + 256 < T_)
      __builtin_prefetch(memb + (size_t)(tc + 256) * (EMB_ / 2), 0, 1);
#pragma unroll 8
    for (int i = 0; i < 256; ++i) {
      const float2 m = memb[(size_t)(tc + i) * (EMB_ / 2)];
      acc.x = fmaf(s_w[i], m.x, acc.x);
      acc.y = fmaf(s_w[i], m.y, acc.y);
    }
  }
  *(float2*)(ctx + b * EMB_ + d) = acc;
}

// ---------------------------------------------------------------------------
extern "C" void kernel_launch(void* const* d_in, const int* in_sizes, int n_in,
                              void* d_out, int out_size, void* d_ws, size_t ws_size,
                              hipStream_t stream) {
  const float* hidden        = (const float*)d_in[0];        // (B, RNN)
  const float* memory        = (const float*)d_in[1];        // (B, T, EMB)
  const float* pm            = (const float*)d_in[2];        // (B, T, ATT)
  const float* awc           = (const float*)d_in[3];        // (B, 2, T)
  const unsigned char* mask  = (const unsigned char*)d_in[4];// (B, T) bool
  const float* Wq            = (const float*)d_in[5];        // (RNN, ATT)
  const float* Wc            = (const float*)d_in[6];        // (32, 2, 31)
  const float* Wlp           = (const float*)d_in[7];        // (32, ATT)
  const float* v             = (const float*)d_in[8];        // (ATT)

  float* out_ctx = (float*)d_out;               // (B, EMB)
  float* out_w   = (float*)d_out + B_ * EMB_;   // (B, T)

  float* ws   = (float*)d_ws;
  float* pq   = ws;                               // B*ATT
  float* loc  = pq + B_ * ATT_;                   // B*T*NF
  float* ener = loc + (size_t)B_ * T_ * NF_;      // B*T

  pq_wmma_kernel<<<dim3(2, 8), 128, 0, stream>>>(hidden, Wq, pq);
  conv_loc_kernel<<<dim3(T_ / 256, B_), 256, 0, stream>>>(awc, Wc, loc);
  energy_wmma_kernel<<<dim3(T_ / 64, B_), 128, 0, stream>>>(loc, Wlp, pq, pm, v, ener);
  softmax_kernel<<<B_, 256, 0, stream>>>(ener, mask, out_w);
  context_kernel<<<dim3(EMB_ / 256, B_), 128, 0, stream>>>(out_w, memory, out_ctx);
}